// DotAttention_84799834292628
// MI455X (gfx1250) — compile-verified
//
#include <hip/hip_runtime.h>
#include <math.h>
#include <stdint.h>

#define B_SZ 32
#define L_SZ 8192
#define D_SZ 128
#define CHUNK 512
#define NCH (L_SZ / CHUNK)          // 16
#define TPW 4                       // 16-row tiles per wave
#define MASK_FILL (-1e25f)

typedef __attribute__((ext_vector_type(2))) float v2f;
typedef __attribute__((ext_vector_type(8))) float v8f;

// Async-copy one 16x128 f32 tile (8 KB, contiguous in global) into LDS.
// 16 x GLOBAL_LOAD_ASYNC_TO_LDS_B128 (each lane moves 16 B) -> ASYNCcnt += 16.
__device__ __forceinline__ void async_tile_load(const float* gsrc, float* ldst, int lane) {
  const uint64_t gbase = (uint64_t)(uintptr_t)gsrc;
  const unsigned lbase = (unsigned)(uintptr_t)ldst;  // low 32 bits = LDS offset
  #pragma unroll
  for (int it = 0; it < 16; ++it) {
    const uint64_t ga = gbase + (uint64_t)((it * 32 + lane) * 16);
    const unsigned la = lbase + (unsigned)((it * 32 + lane) * 16);
    asm volatile("global_load_async_to_lds_b128 %0, %1, off"
                 :: "v"(la), "v"(ga) : "memory");
  }
}

// ---------------------------------------------------------------------------
// Phase 1: raw scores (-> d_out weight area) + online-softmax partials
// (m, s, z[128]) per (batch, chunk) -> workspace.
// grid = B*NCH blocks, 256 threads (8 waves), 4 double-buffered tiles/wave.
// ---------------------------------------------------------------------------
__global__ __launch_bounds__(256) void k_score_partial(
    const float* __restrict__ q,            // [B, 1, D]
    const float* __restrict__ ctx,          // [B, L, D]
    const unsigned char* __restrict__ mask, // [B, L] (bool, 1 byte)
    const int* __restrict__ scale_p,        // [1]
    float* __restrict__ out_w,              // raw scores -> d_out[0 : B*L]
    float* __restrict__ zpart,              // [B*NCH*128]
    float* __restrict__ mpart,              // [B*NCH]
    float* __restrict__ spart)              // [B*NCH]
{
  __shared__ __align__(16) float qlds[D_SZ];
  __shared__ __align__(16) float tile[8][2][16 * D_SZ]; // 2 x 8 KB per wave
  __shared__ float wm[8], wsum[8];
  __shared__ __align__(16) float wz[8][D_SZ];

  const int blk  = blockIdx.x;
  const int b    = blk >> 4;        // / NCH
  const int ch   = blk & (NCH - 1);
  const int tid  = threadIdx.x;
  const int wave = tid >> 5;
  const int lane = tid & 31;

  const float inv_scale = 1.0f / (float)(*scale_p);

  if (tid < D_SZ) qlds[tid] = q[b * D_SZ + tid];
  __syncthreads();

  const int o = (lane >> 4) << 1;   // 0 for lanes 0-15, 2 for lanes 16-31
  const int j = lane & 15;          // output column == row-in-tile

  float  m_run = -INFINITY;
  float  s_run = 0.0f;
  float4 accz  = make_float4(0.f, 0.f, 0.f, 0.f);

  const float* ctxb = ctx + (size_t)b * L_SZ * D_SZ;
  const int tbase = ch * CHUNK + wave * (TPW * 16);

  // prologue: start tile 0
  async_tile_load(ctxb + (size_t)tbase * D_SZ, tile[wave][0], lane);

  for (int tt = 0; tt < TPW; ++tt) {
    const int row0 = tbase + tt * 16;

    // stream next tile into the other buffer while we consume this one
    if (tt + 1 < TPW) {
      async_tile_load(ctxb + (size_t)(row0 + 16) * D_SZ, tile[wave][(tt + 1) & 1], lane);
      asm volatile("s_wait_asynccnt 0x10" ::: "memory"); // current tile's 16 done
    } else {
      asm volatile("s_wait_asynccnt 0x0" ::: "memory");
    }
    const float* tl = tile[wave][tt & 1];

    // ---- scores via fp32 WMMA: D = A(16x4: q bcast) x B(4x16: ctx rows) ----
    // B[k][j] = ctx[row0+j][4t+k] ; A[i][k] = q[4t+k]  => D[i][j] = score_j
    v8f acc = {};
    #pragma unroll
    for (int t = 0; t < 32; ++t) {
      v2f a  = { qlds[4 * t + o],          qlds[4 * t + o + 1] };
      v2f bm = { tl[j * D_SZ + 4 * t + o], tl[j * D_SZ + 4 * t + o + 1] };
      acc = __builtin_amdgcn_wmma_f32_16x16x4_f32(
          /*neg_a=*/false, a, /*neg_b=*/false, bm,
          /*c_mod=*/(short)0, acc, /*reuse_a=*/false, /*reuse_b=*/false);
    }
    float score = acc[0] * inv_scale;       // lane holds score of row0 + j

    const int row = row0 + j;
    if (mask[(size_t)b * L_SZ + row]) score = MASK_FILL;
    if (lane < 16) out_w[(size_t)b * L_SZ + row] = score;   // raw score

    // ---- online softmax update (reduce within each 16-lane half) ----
    float tmax = score;
    #pragma unroll
    for (int mk = 8; mk >= 1; mk >>= 1)
      tmax = fmaxf(tmax, __shfl_xor(tmax, mk, 32));
    const float m_new = fmaxf(m_run, tmax);
    const float corr  = expf(m_run - m_new);
    const float e     = expf(score - m_new);
    float esum = e;
    #pragma unroll
    for (int mk = 8; mk >= 1; mk >>= 1)
      esum += __shfl_xor(esum, mk, 32);
    s_run = s_run * corr + esum;
    accz.x *= corr; accz.y *= corr; accz.z *= corr; accz.w *= corr;

    // ---- z partial: lane owns columns 4*lane..4*lane+3 ----
    #pragma unroll
    for (int r = 0; r < 16; ++r) {
      const float  er = __shfl(e, r, 32);
      const float4 c  = ((const float4*)(tl + r * D_SZ))[lane];
      accz.x += er * c.x; accz.y += er * c.y;
      accz.z += er * c.z; accz.w += er * c.w;
    }
    m_run = m_new;
  }

  if (lane == 0) { wm[wave] = m_run; wsum[wave] = s_run; }
  ((float4*)wz[wave])[lane] = accz;
  __syncthreads();

  // ---- combine 8 waves -> one record per (b, chunk) ----
  if (tid < D_SZ) {
    float M = -INFINITY;
    #pragma unroll
    for (int w = 0; w < 8; ++w) M = fmaxf(M, wm[w]);
    float S = 0.f, Z = 0.f;
    #pragma unroll
    for (int w = 0; w < 8; ++w) {
      const float f = expf(wm[w] - M);
      S += wsum[w] * f;
      Z += wz[w][tid] * f;
    }
    const int rec = b * NCH + ch;
    zpart[rec * D_SZ + tid] = Z;
    if (tid == 0) { mpart[rec] = M; spart[rec] = S; }
  }
}

// ---------------------------------------------------------------------------
// Phase 2: merge chunk partials per batch -> z output + per-batch (M, 1/S).
// grid = B blocks, 128 threads (thread = column d).
// ---------------------------------------------------------------------------
__global__ __launch_bounds__(128) void k_combine(
    const float* __restrict__ zpart,
    const float* __restrict__ mpart,
    const float* __restrict__ spart,
    float* __restrict__ z_out,      // d_out + B*L
    float* __restrict__ bM,
    float* __restrict__ bInvS)
{
  const int b = blockIdx.x;
  const int d = threadIdx.x;
  __shared__ float sm[NCH], ss[NCH];
  if (d < NCH) { sm[d] = mpart[b * NCH + d]; ss[d] = spart[b * NCH + d]; }
  __syncthreads();

  float M = -INFINITY;
  #pragma unroll
  for (int c = 0; c < NCH; ++c) M = fmaxf(M, sm[c]);
  float S = 0.f, Z = 0.f;
  for (int c = 0; c < NCH; ++c) {
    const float f = expf(sm[c] - M);
    S += ss[c] * f;
    Z += zpart[(b * NCH + c) * D_SZ + d] * f;
  }
  z_out[b * D_SZ + d] = Z / S;
  if (d == 0) { bM[b] = M; bInvS[b] = 1.0f / S; }
}

// ---------------------------------------------------------------------------
// Phase 3: normalize raw scores in place -> softmax weights.
// ---------------------------------------------------------------------------
__global__ __launch_bounds__(256) void k_norm_w(
    float* __restrict__ w,
    const float* __restrict__ bM,
    const float* __restrict__ bInvS)
{
  const int idx = blockIdx.x * 256 + threadIdx.x;  // < B*L
  const int b = idx >> 13;                         // / 8192
  const float raw = w[idx];
  w[idx] = expf(raw - bM[b]) * bInvS[b];           // masked: exp(-1e25-M)=0
}

// ---------------------------------------------------------------------------
extern "C" void kernel_launch(void* const* d_in, const int* in_sizes, int n_in,
                              void* d_out, int out_size, void* d_ws, size_t ws_size,
                              hipStream_t stream) {
  const float*         q     = (const float*)d_in[0];
  const float*         ctx   = (const float*)d_in[1];
  const unsigned char* mask  = (const unsigned char*)d_in[2];
  const int*           scale = (const int*)d_in[3];

  float* out_w = (float*)d_out;                       // [B*L] weights
  float* out_z = (float*)d_out + (size_t)B_SZ * L_SZ; // [B*D] z

  float* wsf   = (float*)d_ws;
  float* zpart = wsf;                                 // B*NCH*128
  float* mpart = zpart + B_SZ * NCH * D_SZ;           // B*NCH
  float* spart = mpart + B_SZ * NCH;                  // B*NCH
  float* bM    = spart + B_SZ * NCH;                  // B
  float* bInvS = bM + B_SZ;                           // B

  k_score_partial<<<B_SZ * NCH, 256, 0, stream>>>(q, ctx, mask, scale,
                                                  out_w, zpart, mpart, spart);
  k_combine<<<B_SZ, 128, 0, stream>>>(zpart, mpart, spart, out_z, bM, bInvS);
  k_norm_w<<<(B_SZ * L_SZ) / 256, 256, 0, stream>>>(out_w, bM, bInvS);
}